// RegionAttention_44435731644833
// MI455X (gfx1250) — compile-verified
//
#include <hip/hip_runtime.h>
#include <stdint.h>

#define TPB 256

typedef __attribute__((ext_vector_type(4))) float v4f;
typedef __attribute__((ext_vector_type(4))) int v4i;

#define AS1 __attribute__((address_space(1)))
#define AS3 __attribute__((address_space(3)))

// ---- CDNA5 feature detection -------------------------------------------------
#if defined(__has_builtin)
#if __has_builtin(__builtin_amdgcn_global_load_async_to_lds_b128)
#define HAVE_ASYNC_LDS 1
#endif
#if __has_builtin(__builtin_amdgcn_s_wait_asynccnt)
#define WAIT_ASYNC(N) __builtin_amdgcn_s_wait_asynccnt(N)
#endif
#endif
#ifndef WAIT_ASYNC
#define WAIT_ASYNC(N) asm volatile("s_wait_asynccnt %0" ::"i"(N) : "memory")
#endif

// ---- Kernel 1: out[i] = 1.0f -------------------------------------------------
__global__ __launch_bounds__(TPB) void ra_init_ones(float* __restrict__ out, int n) {
  int i = blockIdx.x * TPB + threadIdx.x;
  if (i < n) out[i] = 1.0f;
}

// ---- bin + blend for one landmark -------------------------------------------
__device__ __forceinline__ void ra_scatter_one(float x, float y,
                                               const float* __restrict__ enh,
                                               float* __restrict__ out) {
  // patch is 16x16 -> multiply by 2^-4 is exact, truncation == floor (x,y >= 0)
  int c = (int)(x * 0.0625f); c = c > 511 ? 511 : c;
  int r = (int)(y * 0.0625f); r = r > 511 ? 511 : r;
  int b = (r << 9) | c;
  // bit-exact replication of: 1 + (e - 1) * mask  with mask == 1
  out[b] = 1.0f + (enh[b] - 1.0f);
}

// ---- Kernel 2: stream landmarks, scatter blended weight ----------------------
// q = number of float4 packets (each = 2 landmarks)
__global__ __launch_bounds__(TPB) void ra_scatter(const v4f* __restrict__ lm4,
                                                  const float* __restrict__ enh,
                                                  float* __restrict__ out, int q) {
#if defined(HAVE_ASYNC_LDS)
  // Double-buffered global->LDS async pipeline (ASYNCcnt data path).
  // Each thread stages 16B per stage; each thread consumes only its own bytes,
  // so per-wave ASYNCcnt ordering suffices and no workgroup barrier is needed.
  __shared__ v4f buf[2][TPB];
  const int tid = threadIdx.x;
  const int nStages = (q + TPB - 1) / TPB;
  const int stride = gridDim.x;
  int s = blockIdx.x;
  if (s >= nStages) return;
  {
    int idx = s * TPB + tid;
    if (idx < q)
      __builtin_amdgcn_global_load_async_to_lds_b128(
          (AS1 v4i*)(lm4 + idx),
          (AS3 v4i*)&buf[0][tid],
          /*offset=*/0, /*cpol=*/0);
  }
  int parity = 0;
  for (; s < nStages; s += stride) {
    int snext = s + stride;
    if (snext < nStages) {
      int idx = snext * TPB + tid;
      if (idx < q)
        __builtin_amdgcn_global_load_async_to_lds_b128(
            (AS1 v4i*)(lm4 + idx),
            (AS3 v4i*)&buf[parity ^ 1][tid],
            /*offset=*/0, /*cpol=*/0);
      WAIT_ASYNC(1);  // oldest (current stage) transfer has landed
    } else {
      WAIT_ASYNC(0);  // drain
    }
    int idx = s * TPB + tid;
    if (idx < q) {
      v4f v = buf[parity][tid];  // ds_load_b128
      ra_scatter_one(v.x, v.y, enh, out);
      ra_scatter_one(v.z, v.w, enh, out);
    }
    parity ^= 1;
  }
#else
  // Fallback: direct non-temporal B128 streaming loads (stream-once data,
  // keep it out of the way of the L2-resident scatter region).
  const int stride = gridDim.x * TPB;
  for (int idx = blockIdx.x * TPB + threadIdx.x; idx < q; idx += stride) {
    v4f v = __builtin_nontemporal_load(&lm4[idx]);
    ra_scatter_one(v.x, v.y, enh, out);
    ra_scatter_one(v.z, v.w, enh, out);
  }
#endif
}

// ---- Host launcher -----------------------------------------------------------
extern "C" void kernel_launch(void* const* d_in, const int* in_sizes, int n_in,
                              void* d_out, int out_size, void* d_ws, size_t ws_size,
                              hipStream_t stream) {
  const float* landmarks = (const float*)d_in[0];  // (L, 2) f32, flat = 2L
  const float* enh       = (const float*)d_in[1];  // (N_BINS,) f32
  float* out             = (float*)d_out;          // (N_BINS,) f32
  (void)d_ws; (void)ws_size; (void)n_in;

  const int nbins = out_size;          // 262144
  const int q     = in_sizes[0] / 4;   // float4 packets (2 landmarks each)

  // Pass 1: default weight everywhere (ordered before scatter on `stream`).
  int initBlocks = (nbins + TPB - 1) / TPB;
  ra_init_ones<<<initBlocks, TPB, 0, stream>>>(out, nbins);

  // Pass 2: stream 67 MB of landmarks, scatter blended weights (L2-resident).
  int nStages = (q + TPB - 1) / TPB;
  int blocks  = nStages < 2048 ? nStages : 2048;  // persistent-ish, ~8 stages/block
  if (blocks < 1) blocks = 1;
  ra_scatter<<<blocks, TPB, 0, stream>>>((const v4f*)landmarks, enh, out, q);
}